// ExportableLocalTransformerCore_30442728194394
// MI455X (gfx1250) — compile-verified
//
#include <hip/hip_runtime.h>
#include <math.h>

// ---------------- problem dims ----------------
#define LAYERS 4
#define BATCH  4
#define SEQ    1024
#define HDIM   1024
#define NHEAD  16
#define HEADD  64
#define FFDIM  4096
#define TOK    (BATCH * SEQ)          // 4096 tokens
#define ATT_SCALE 0.125f              // 1/sqrt(64)
#define NEGF   (-3.4028234663852886e38f)

typedef float v2f __attribute__((ext_vector_type(2)));
typedef float v8f __attribute__((ext_vector_type(8)));

// fp32 WMMA: D(16x16,f32) = A(16x4,f32) x B(4x16,f32) + C
__device__ __forceinline__ v8f wmma_f32(v2f a, v2f b, v8f c) {
    return __builtin_amdgcn_wmma_f32_16x16x4_f32(
        /*neg_a=*/false, a, /*neg_b=*/false, b,
        /*c_mod=*/(short)0, c, /*reuse_a=*/false, /*reuse_b=*/false);
}

__device__ __forceinline__ float gelu_tanh(float x) {
    float x3 = x * x * x;
    return 0.5f * x * (1.0f + tanhf(0.7978845608028654f * (x + 0.044715f * x3)));
}

// ---------------- mask / position ids ----------------
__global__ void mask_pid_kernel(const unsigned char* __restrict__ mask,
                                float* __restrict__ maskf, int* __restrict__ pid) {
    int b = blockIdx.x, s = threadIdx.x;
    int c = 0;
    for (int i = 0; i <= s; ++i) c += mask[b * SEQ + i] ? 1 : 0;
    int idx = b * SEQ + s;
    int m = mask[idx] ? 1 : 0;
    maskf[idx] = (float)m;
    pid[idx] = m ? (c - 1) : 0;
}

// ---------------- h = embeds * maskf ----------------
__global__ void scale_embed_kernel(const float* __restrict__ e,
                                   const float* __restrict__ maskf,
                                   float* __restrict__ h) {
    int i = blockIdx.x * 256 + threadIdx.x;
    h[i] = e[i] * maskf[i >> 10];   // HDIM == 1024
}

// ---------------- LayerNorm (block per row) ----------------
__global__ __launch_bounds__(256) void ln_kernel(const float* __restrict__ x,
                                                 const float* __restrict__ g,
                                                 const float* __restrict__ bta,
                                                 float* __restrict__ y,
                                                 const float* __restrict__ rowmask) {
    __shared__ float red[256];
    int row = blockIdx.x, tid = threadIdx.x;
    const float* xr = x + (size_t)row * HDIM;
    float v[4];
    float s = 0.f;
#pragma unroll
    for (int i = 0; i < 4; ++i) { v[i] = xr[tid + i * 256]; s += v[i]; }
    red[tid] = s; __syncthreads();
    for (int off = 128; off > 0; off >>= 1) {
        if (tid < off) red[tid] += red[tid + off];
        __syncthreads();
    }
    float mean = red[0] * (1.0f / HDIM);
    __syncthreads();
    float s2 = 0.f;
#pragma unroll
    for (int i = 0; i < 4; ++i) { float d = v[i] - mean; s2 += d * d; }
    red[tid] = s2; __syncthreads();
    for (int off = 128; off > 0; off >>= 1) {
        if (tid < off) red[tid] += red[tid + off];
        __syncthreads();
    }
    float rstd = rsqrtf(red[0] * (1.0f / HDIM) + 1e-5f);
    float mk = rowmask ? rowmask[row] : 1.0f;
#pragma unroll
    for (int i = 0; i < 4; ++i) {
        int c = tid + i * 256;
        y[(size_t)row * HDIM + c] = ((v[i] - mean) * rstd * g[c] + bta[c]) * mk;
    }
}

// ---------------- fp32 WMMA GEMM ----------------
// C[M,N] = act(A[M,K] @ W[K,N] + bias) (+resid) (*rowmask)
// block tile 64x256, 8 waves, each wave 32x64 (2x4 of 16x16 WMMA tiles).
// LDS holds K-pair-interleaved float2 so every WMMA operand is one ds_load_b64.
// Row strides are 32 dwords mod 64 so the two 16-lane halves (reading kp and
// kp+1) hit disjoint LDS bank sets.
__global__ __launch_bounds__(256) void gemm_wmma_kernel(
        const float* __restrict__ A, const float* __restrict__ W,
        const float* __restrict__ bias, const float* __restrict__ resid,
        const float* __restrict__ rowmask, float* __restrict__ C,
        int M, int N, int K, int act) {
    __shared__ float2 As2[8][80];    // [kpair][m]  (pair = {A[m][2kp],A[m][2kp+1]})
    __shared__ float2 Bs2[8][272];   // [kpair][n]
    const int tid = threadIdx.x;
    const int wave = tid >> 5, lane = tid & 31;
    const int lh = lane >> 4, ll = lane & 15;
    const int m0 = blockIdx.y * 64, n0 = blockIdx.x * 256;
    const int wm = (wave & 1) * 32, wn = (wave >> 1) * 64;

    v8f acc[2][4];
#pragma unroll
    for (int mt = 0; mt < 2; ++mt)
#pragma unroll
        for (int nt = 0; nt < 4; ++nt)
#pragma unroll
            for (int r = 0; r < 8; ++r) acc[mt][nt][r] = 0.0f;

    const int am = tid >> 2, akq = (tid & 3) * 4;  // A: 64 rows x 16 k, float4 each
    const int bkp = tid >> 5, bc0 = (tid & 31) * 4; // B: pair row per wave

    for (int k0 = 0; k0 < K; k0 += 16) {
        // stage A (64x16) as k-pairs
        {
            float4 av = *(const float4*)(A + (size_t)(m0 + am) * K + k0 + akq);
            As2[(akq >> 1) + 0][am] = make_float2(av.x, av.y);
            As2[(akq >> 1) + 1][am] = make_float2(av.z, av.w);
        }
        // stage B (16x256) as k-pairs: wave w handles pair row w
        {
            const float* r0 = W + (size_t)(k0 + 2 * bkp) * N + n0;
            const float* r1 = r0 + N;
#pragma unroll
            for (int p = 0; p < 2; ++p) {
                int c = p * 128 + bc0;
                float4 lo = *(const float4*)(r0 + c);
                float4 hi = *(const float4*)(r1 + c);
                Bs2[bkp][c + 0] = make_float2(lo.x, hi.x);
                Bs2[bkp][c + 1] = make_float2(lo.y, hi.y);
                Bs2[bkp][c + 2] = make_float2(lo.z, hi.z);
                Bs2[bkp][c + 3] = make_float2(lo.w, hi.w);
            }
        }
        __syncthreads();
#pragma unroll
        for (int kk = 0; kk < 16; kk += 4) {
            int kp = (kk >> 1) + lh;   // lane half picks K{kk,kk+1} or K{kk+2,kk+3}
            v2f a[2], bf[4];
            a[0] = *(const v2f*)&As2[kp][wm + ll];
            a[1] = *(const v2f*)&As2[kp][wm + 16 + ll];
#pragma unroll
            for (int nt = 0; nt < 4; ++nt)
                bf[nt] = *(const v2f*)&Bs2[kp][wn + nt * 16 + ll];
#pragma unroll
            for (int mt = 0; mt < 2; ++mt)
#pragma unroll
                for (int nt = 0; nt < 4; ++nt)
                    acc[mt][nt] = wmma_f32(a[mt], bf[nt], acc[mt][nt]);
        }
        __syncthreads();
    }

#pragma unroll
    for (int mt = 0; mt < 2; ++mt)
#pragma unroll
        for (int nt = 0; nt < 4; ++nt)
#pragma unroll
            for (int r = 0; r < 8; ++r) {
                int row = m0 + wm + mt * 16 + r + 8 * lh;
                int col = n0 + wn + nt * 16 + ll;
                float v = acc[mt][nt][r] + bias[col];
                if (act) v = gelu_tanh(v);
                size_t idx = (size_t)row * N + col;
                if (resid) v += resid[idx];
                if (rowmask) v *= rowmask[row];
                C[idx] = v;
            }
}

// ---------------- RoPE + QKV split (to head-major [B,NH,S,HD]) ----------------
__global__ void rope_split_kernel(const float* __restrict__ qkv,
                                  const int* __restrict__ pid,
                                  float* __restrict__ qh, float* __restrict__ kh,
                                  float* __restrict__ vh) {
    int idx = blockIdx.x * 256 + threadIdx.x;   // TOK*NHEAD*32 threads (one pair)
    int d2 = idx & 31;
    int h  = (idx >> 5) & 15;
    int t  = idx >> 9;            // token 0..4095
    int b = t >> 10, s = t & 1023;
    const float* base = qkv + (size_t)t * (3 * HDIM);
    int co = h * HEADD + 2 * d2;
    float qe = base[co],            qo = base[co + 1];
    float ke = base[HDIM + co],     ko = base[HDIM + co + 1];
    float ve = base[2 * HDIM + co], vo = base[2 * HDIM + co + 1];
    float p = (float)pid[t];
    float inv = __powf(10000.0f, -((float)(2 * d2) / (float)HEADD));
    float ang = p * inv;
    float sn, cs;
    __sincosf(ang, &sn, &cs);
    size_t o = ((size_t)(b * NHEAD + h) * SEQ + s) * HEADD + 2 * d2;
    qh[o] = qe * cs - qo * sn;  qh[o + 1] = qo * cs + qe * sn;
    kh[o] = ke * cs - ko * sn;  kh[o + 1] = ko * cs + ke * sn;
    vh[o] = ve;                 vh[o + 1] = vo;
}

// ---------------- flash-style causal attention (fp32 WMMA) ----------------
// grid (SEQ/128, NHEAD, BATCH), 256 threads = 8 waves, wave owns 16 q rows
__global__ __launch_bounds__(256) void attn_wmma_kernel(
        const float* __restrict__ qh, const float* __restrict__ kh,
        const float* __restrict__ vh, const float* __restrict__ maskf,
        float* __restrict__ out) {
    __shared__ float  Ks[16][68];    // [key][d] (row-major, pairs contiguous in d)
    __shared__ float2 Vp[8][80];     // [keypair][d] = {V[2kp][d], V[2kp+1][d]}
    __shared__ float  Ps[8][16][18]; // per-wave P patch for re-fragmenting
    __shared__ float  Ms[16];
    const int tid = threadIdx.x;
    const int wave = tid >> 5, lane = tid & 31;
    const int lh = lane >> 4, ll = lane & 15;
    const int qc = blockIdx.x, hh = blockIdx.y, b = blockIdx.z;
    const int q0 = qc * 128 + wave * 16;
    const size_t headoff = (size_t)(b * NHEAD + hh) * SEQ * HEADD;
    const float* Qb = qh + headoff;
    const float* Kb = kh + headoff;
    const float* Vb = vh + headoff;

    // preload Q as 16 A-fragments (K dim = 64 -> 16 steps of 4)
    v2f qf[16];
    {
        const float* qr = Qb + (size_t)(q0 + ll) * HEADD + 2 * lh;
#pragma unroll
        for (int kk = 0; kk < 16; ++kk) {
            float2 t = *(const float2*)(qr + kk * 4);
            qf[kk].x = t.x; qf[kk].y = t.y;
        }
    }

    v8f o[4];
#pragma unroll
    for (int nt = 0; nt < 4; ++nt)
#pragma unroll
        for (int r = 0; r < 8; ++r) o[nt][r] = 0.0f;
    float mrow[8], lrow[8];
#pragma unroll
    for (int r = 0; r < 8; ++r) { mrow[r] = -INFINITY; lrow[r] = 0.0f; }

    const int nkt = (qc + 1) * 8;  // causal: keys up to end of this q-chunk
    for (int kt = 0; kt < nkt; ++kt) {
        const int k0 = kt * 16;
        {   // cooperative stage of K tile (row-major) + V tile (pair-interleaved)
            int r = tid >> 4, c4 = (tid & 15) * 4;
            *(float4*)&Ks[r][c4] = *(const float4*)(Kb + (size_t)(k0 + r) * HEADD + c4);
            int kp = tid >> 5, d0 = (tid & 31) * 2;
            float2 lo = *(const float2*)(Vb + (size_t)(k0 + 2 * kp) * HEADD + d0);
            float2 hi = *(const float2*)(Vb + (size_t)(k0 + 2 * kp + 1) * HEADD + d0);
            Vp[kp][d0]     = make_float2(lo.x, hi.x);
            Vp[kp][d0 + 1] = make_float2(lo.y, hi.y);
            if (tid < 16) Ms[tid] = maskf[b * SEQ + k0 + tid];
        }
        __syncthreads();
        if (k0 <= q0 + 15) {   // wave-uniform causal skip
            v8f sc;
#pragma unroll
            for (int r = 0; r < 8; ++r) sc[r] = 0.0f;
#pragma unroll
            for (int kk = 0; kk < 16; ++kk) {
                v2f bf = *(const v2f*)&Ks[ll][kk * 4 + 2 * lh]; // B[d][n]=K[k0+n][d]
                sc = wmma_f32(qf[kk], bf, sc);
            }
            float mk = Ms[ll];
            int kidx = k0 + ll;
            float p[8], fac[8];
#pragma unroll
            for (int r = 0; r < 8; ++r) {
                int qidx = q0 + r + 8 * lh;
                float v = sc[r] * ATT_SCALE;
                if (kidx > qidx || mk == 0.0f) v = NEGF;
                float rm = v;                       // row max over 16-lane half
                rm = fmaxf(rm, __shfl_xor(rm, 1));
                rm = fmaxf(rm, __shfl_xor(rm, 2));
                rm = fmaxf(rm, __shfl_xor(rm, 4));
                rm = fmaxf(rm, __shfl_xor(rm, 8));
                float mnew = fmaxf(mrow[r], rm);
                float f = __expf(mrow[r] - mnew);
                float pe = __expf(v - mnew);
                float rs = pe;
                rs += __shfl_xor(rs, 1);
                rs += __shfl_xor(rs, 2);
                rs += __shfl_xor(rs, 4);
                rs += __shfl_xor(rs, 8);
                lrow[r] = lrow[r] * f + rs;
                mrow[r] = mnew; fac[r] = f; p[r] = pe;
            }
#pragma unroll
            for (int nt = 0; nt < 4; ++nt)
#pragma unroll
                for (int r = 0; r < 8; ++r) o[nt][r] *= fac[r];
            // re-fragment P (C layout -> A layout) via per-wave LDS patch
#pragma unroll
            for (int r = 0; r < 8; ++r) Ps[wave][r + 8 * lh][ll] = p[r];
#pragma unroll
            for (int kk = 0; kk < 4; ++kk) {
                v2f a = *(const v2f*)&Ps[wave][ll][kk * 4 + 2 * lh];
                int kp = kk * 2 + lh;   // V key-pair index
#pragma unroll
                for (int nt = 0; nt < 4; ++nt) {
                    v2f bf = *(const v2f*)&Vp[kp][nt * 16 + ll];
                    o[nt] = wmma_f32(a, bf, o[nt]);
                }
            }
        }
        __syncthreads();
    }

#pragma unroll
    for (int r = 0; r < 8; ++r) {
        int qidx = q0 + r + 8 * lh;
        float inv = 1.0f / lrow[r];
        size_t base = (size_t)(b * SEQ + qidx) * HDIM + hh * HEADD;
#pragma unroll
        for (int nt = 0; nt < 4; ++nt)
            out[base + nt * 16 + ll] = o[nt][r] * inv;
    }
}

// ---------------- launcher ----------------
extern "C" void kernel_launch(void* const* d_in, const int* in_sizes, int n_in,
                              void* d_out, int out_size, void* d_ws, size_t ws_size,
                              hipStream_t stream) {
    const float* emb           = (const float*)d_in[0];
    const unsigned char* amask = (const unsigned char*)d_in[1];
    const float* Wqkv = (const float*)d_in[2];
    const float* bqkv = (const float*)d_in[3];
    const float* Wo   = (const float*)d_in[4];
    const float* bo   = (const float*)d_in[5];
    const float* ln1g = (const float*)d_in[6];
    const float* ln1b = (const float*)d_in[7];
    const float* ln2g = (const float*)d_in[8];
    const float* ln2b = (const float*)d_in[9];
    const float* Wfc  = (const float*)d_in[10];
    const float* bfc  = (const float*)d_in[11];
    const float* Wproj = (const float*)d_in[12];
    const float* bproj = (const float*)d_in[13];
    const float* lnfg = (const float*)d_in[14];
    const float* lnfb = (const float*)d_in[15];

    float* ws    = (float*)d_ws;
    float* maskf = ws;                                   // 4096
    int*   pid   = (int*)(ws + 4096);                    // 4096
    float* hbuf  = ws + 8192;                            // TOK*HDIM
    float* lno   = hbuf + (size_t)TOK * HDIM;            // TOK*HDIM
    float* big   = lno + (size_t)TOK * HDIM;             // TOK*FFDIM (qkv & ff reuse)
    float* qbuf  = big + (size_t)TOK * FFDIM;            // TOK*HDIM
    float* kbuf  = qbuf + (size_t)TOK * HDIM;
    float* vbuf  = kbuf + (size_t)TOK * HDIM;
    float* attnb = vbuf + (size_t)TOK * HDIM;

    mask_pid_kernel<<<BATCH, SEQ, 0, stream>>>(amask, maskf, pid);
    scale_embed_kernel<<<(TOK * HDIM) / 256, 256, 0, stream>>>(emb, maskf, hbuf);

    for (int l = 0; l < LAYERS; ++l) {
        // ln1
        ln_kernel<<<TOK, 256, 0, stream>>>(hbuf, ln1g + l * HDIM, ln1b + l * HDIM,
                                           lno, nullptr);
        // qkv = ln1(h) @ Wqkv + bqkv
        gemm_wmma_kernel<<<dim3(3 * HDIM / 256, TOK / 64), 256, 0, stream>>>(
            lno, Wqkv + (size_t)l * HDIM * 3 * HDIM, bqkv + (size_t)l * 3 * HDIM,
            nullptr, nullptr, big, TOK, 3 * HDIM, HDIM, 0);
        // rope + split to head-major q/k/v
        rope_split_kernel<<<(TOK * NHEAD * 32) / 256, 256, 0, stream>>>(
            big, pid, qbuf, kbuf, vbuf);
        // attention
        attn_wmma_kernel<<<dim3(SEQ / 128, NHEAD, BATCH), 256, 0, stream>>>(
            qbuf, kbuf, vbuf, maskf, attnb);
        // h = h + attn @ Wo + bo
        gemm_wmma_kernel<<<dim3(HDIM / 256, TOK / 64), 256, 0, stream>>>(
            attnb, Wo + (size_t)l * HDIM * HDIM, bo + (size_t)l * HDIM,
            hbuf, nullptr, hbuf, TOK, HDIM, HDIM, 0);
        // ln2
        ln_kernel<<<TOK, 256, 0, stream>>>(hbuf, ln2g + l * HDIM, ln2b + l * HDIM,
                                           lno, nullptr);
        // ff = gelu(ln2(h) @ Wfc + bfc)
        gemm_wmma_kernel<<<dim3(FFDIM / 256, TOK / 64), 256, 0, stream>>>(
            lno, Wfc + (size_t)l * HDIM * FFDIM, bfc + (size_t)l * FFDIM,
            nullptr, nullptr, big, TOK, FFDIM, HDIM, 1);
        // h = (h + ff @ Wproj + bproj) * maskf
        gemm_wmma_kernel<<<dim3(HDIM / 256, TOK / 64), 256, 0, stream>>>(
            big, Wproj + (size_t)l * FFDIM * HDIM, bproj + (size_t)l * HDIM,
            hbuf, maskf, hbuf, TOK, HDIM, FFDIM, 0);
    }
    // final ln * maskf -> d_out
    ln_kernel<<<TOK, 256, 0, stream>>>(hbuf, lnfg, lnfb, (float*)d_out, maskf);
}